// CoPE_33432025432167
// MI455X (gfx1250) — compile-verified
//
#include <hip/hip_runtime.h>
#include <hip/hip_bf16.h>

typedef float v2f __attribute__((ext_vector_type(2)));
typedef float v8f __attribute__((ext_vector_type(8)));

// ---------------------------------------------------------------------------
// Kernel 1: logits_int[M,64] = Q[M,64] @ P[64,64] with V_WMMA_F32_16X16X4_F32
// One wave computes a 16(M) x 64(N) tile: 16 K-steps x 4 N-tiles.
// A layout (ISA 7.12.2, 32-bit A 16x4): lane m=lane&15; lanes<16 hold K={k,k+1},
// lanes>=16 hold K={k+2,k+3} in the VGPR pair.
// B layout (4x16, rows striped across lanes like C/D): lane n=lane&15;
// VGPR0 holds K=k (lanes 0-15) / K=k+2 (lanes 16-31), VGPR1 holds K=k+1 / k+3.
// C/D layout: VGPR r: lanes 0-15 -> M=r, lanes 16-31 -> M=r+8, N=lane&15.
// ---------------------------------------------------------------------------
__global__ void __launch_bounds__(256) cope_gemm_wmma(
    const float* __restrict__ Q,   // [rows, 64]
    const float* __restrict__ P,   // [64, 64] (d-major, n contiguous)
    float* __restrict__ L) {       // [rows, 64]
  const int lane = threadIdx.x & 31;
  const int wv   = threadIdx.x >> 5;
  const int tile = blockIdx.x * 8 + wv;          // 16-row tile index
  const int n    = lane & 15;                    // N (for B/C/D); also M for A
  const int kk   = (lane >> 4) << 1;             // 0 or 2 (half-wave K offset)

  const float* qrow = Q + (size_t)(tile * 16 + n) * 64;

  v8f c0 = {}, c1 = {}, c2 = {}, c3 = {};
#pragma unroll
  for (int s = 0; s < 16; ++s) {
    const int k = s * 4 + kk;
    v2f a;
    a.x = qrow[k];
    a.y = qrow[k + 1];
    const float* p = P + k * 64 + n;
    v2f b0; b0.x = p[0];   b0.y = p[64];
    v2f b1; b1.x = p[16];  b1.y = p[80];
    v2f b2; b2.x = p[32];  b2.y = p[96];
    v2f b3; b3.x = p[48];  b3.y = p[112];
    c0 = __builtin_amdgcn_wmma_f32_16x16x4_f32(false, a, false, b0, (short)0, c0, false, false);
    c1 = __builtin_amdgcn_wmma_f32_16x16x4_f32(false, a, false, b1, (short)0, c1, false, false);
    c2 = __builtin_amdgcn_wmma_f32_16x16x4_f32(false, a, false, b2, (short)0, c2, false, false);
    c3 = __builtin_amdgcn_wmma_f32_16x16x4_f32(false, a, false, b3, (short)0, c3, false, false);
  }

  const int mh = (lane >> 4) * 8;
#pragma unroll
  for (int r = 0; r < 8; ++r) {
    float* o = L + (size_t)(tile * 16 + mh + r) * 64 + n;
    o[0]  = c0[r];
    o[16] = c1[r];
    o[32] = c2[r];
    o[48] = c3[r];
  }
}

// ---------------------------------------------------------------------------
// Kernel 2: per-row reverse cumsum of sigmoid(attn_logits) + interpolation.
// One wave per row; 128-element blocks processed back-to-front with coalesced
// float4 loads/stores; wave-level suffix scan via __shfl_down; 64-float logits
// row held in a per-wave LDS slice for the indexed gather.
// ---------------------------------------------------------------------------
__device__ __forceinline__ float fast_sigmoid(float x) {
  // 1 / (1 + exp(-x)) via v_exp_f32 (exp2) + v_rcp_f32
  const float e = __builtin_amdgcn_exp2f(-x * 1.4426950408889634f);
  return __builtin_amdgcn_rcpf(1.0f + e);
}

__device__ __forceinline__ float cope_interp(float p, const float* l) {
  p = fminf(p, 63.0f);                 // clip(a_max = NPOS_MAX-1); p >= 0 always
  const float pf = floorf(p);
  const int   fi = (int)pf;
  const int   ci = (int)ceilf(p);
  const float w  = p - pf;
  return l[ci] * w + l[fi] * (1.0f - w);
}

template <bool PRECOMP>
__global__ void __launch_bounds__(256) cope_scan_kernel(
    const float* __restrict__ A,     // attn_logits [rows, S]
    const float* __restrict__ Lsrc,  // logits_int  [rows, 64] (PRECOMP path)
    const float* __restrict__ Q,     // [rows, 64]  (fallback path)
    const float* __restrict__ P,     // [64, 64]    (fallback path)
    float* __restrict__ O,           // [rows, S]
    int S) {
  __shared__ float lds[8][64];
  const int lane = threadIdx.x & 31;
  const int wv   = threadIdx.x >> 5;
  const long long row = (long long)blockIdx.x * 8 + wv;
  float* lrow = &lds[wv][0];

  if (PRECOMP) {
    lrow[lane]      = Lsrc[row * 64 + lane];
    lrow[lane + 32] = Lsrc[row * 64 + 32 + lane];
  } else {
    // fallback: per-row scalar dot (no scratch available)
    const float* qrow = Q + row * 64;
    float a0 = 0.f, a1 = 0.f;
#pragma unroll
    for (int d = 0; d < 64; ++d) {
      const float qv = qrow[d];
      a0 = fmaf(qv, P[d * 64 + lane],      a0);
      a1 = fmaf(qv, P[d * 64 + lane + 32], a1);
    }
    lrow[lane]      = a0;
    lrow[lane + 32] = a1;
  }
  // LDS ops are in-order within a wave; each wave touches only its own slice.

  const float4* a4 = (const float4*)(A + row * (long long)S);
  float4*       o4 = (float4*)(O + row * (long long)S);

  float carry = 0.0f;                      // suffix sum of all later blocks
  for (int blk = (S >> 7) - 1; blk >= 0; --blk) {
    const float4 g = a4[blk * 32 + lane];  // elements [blk*128 + 4*lane, +4)
    const float g0 = fast_sigmoid(g.x);
    const float g1 = fast_sigmoid(g.y);
    const float g2 = fast_sigmoid(g.z);
    const float g3 = fast_sigmoid(g.w);

    // lane-local suffix sums (element j includes gates j..3 of this lane)
    const float s3 = g3;
    const float s2 = s3 + g2;
    const float s1 = s2 + g1;
    const float s0 = s1 + g0;              // lane total

    // wave-level inclusive suffix scan of lane totals
    float incl = s0;
#pragma unroll
    for (int off = 1; off < 32; off <<= 1) {
      const float v = __shfl_down(incl, off, 32);
      incl += (lane + off < 32) ? v : 0.0f;
    }
    const float total = __shfl(incl, 0, 32);   // sum of whole 128-block
    const float base  = carry + (incl - s0);   // suffix-exclusive across lanes

    float4 o;
    o.x = cope_interp(base + s0, lrow);
    o.y = cope_interp(base + s1, lrow);
    o.z = cope_interp(base + s2, lrow);
    o.w = cope_interp(base + s3, lrow);
    o4[blk * 32 + lane] = o;

    carry += total;
  }
}

extern "C" void kernel_launch(void* const* d_in, const int* in_sizes, int n_in,
                              void* d_out, int out_size, void* d_ws, size_t ws_size,
                              hipStream_t stream) {
  (void)n_in; (void)out_size;
  const float* Q = (const float*)d_in[0];   // [B,S,64]
  const float* A = (const float*)d_in[1];   // [B,S,S]
  const float* P = (const float*)d_in[2];   // [1,64,64]
  float* O = (float*)d_out;                 // [B,S,S]

  const long long rows = (long long)in_sizes[0] / 64;       // B*S = 32768
  const int S = (int)((long long)in_sizes[1] / rows);       // 1024

  const size_t lbytes = (size_t)rows * 64 * sizeof(float);  // 8 MB scratch
  const int scanBlocks = (int)(rows / 8);                   // 8 rows per block

  if (ws_size >= lbytes && (rows % 128) == 0) {
    float* L = (float*)d_ws;
    cope_gemm_wmma<<<(int)(rows / 128), 256, 0, stream>>>(Q, P, L);
    cope_scan_kernel<true><<<scanBlocks, 256, 0, stream>>>(A, L, Q, P, O, S);
  } else {
    cope_scan_kernel<false><<<scanBlocks, 256, 0, stream>>>(A, nullptr, Q, P, O, S);
  }
}